// HybridLoss_42863773614450
// MI455X (gfx1250) — compile-verified
//
#include <hip/hip_runtime.h>
#include <hip/hip_bf16.h>
#include <stdint.h>

// ---------------------------------------------------------------------------
// HybridLoss on (4,1,160,160,160) fp32.
//   pooled = separable 31^3 box sum (3 passes, zero pad), /31^3 fused later.
//   pass1: pool along H (stride 160)      target -> bufA   (running sum)
//   pass2: pool along D (stride 25600)    bufA   -> bufB   (running sum)
//   pass3: pool along W (stride 1)        bufB   -> bufA   (async->LDS + prefix)
//   pass4: fused elementwise + per-(b,chunk) partial reductions (6 sums)
//   pass5: finalize — WMMA f32 16x16x4 tree-sums 512 partials per (b,q),
//          computes total_loss / dice / iou -> d_out[3]
// ---------------------------------------------------------------------------

#define DIMB 4
#define DIML 160
#define PLANE (DIML * DIML)            // 25600
#define VOL   (DIML * DIML * DIML)     // 4096000 per batch
#define NTOT  (DIMB * VOL)             // 16384000
#define KRAD  15
#define KVOL  29791.0f                 // 31^3
#define CHUNKS_PER_B 512
#define CHUNK_ELEMS  8000              // 512*8000 == VOL
#define NQ 6                           // weit, iw, uw, i, u, bce

typedef __attribute__((ext_vector_type(2))) float v2f;
typedef __attribute__((ext_vector_type(8))) float v8f;

// --------------------------- strided sliding-sum ---------------------------
// One thread owns one line of 160 elements along the pooled axis (element
// stride `stride`). Lanes are contiguous in W -> coalesced; data is L2
// resident after the first pass. Running sum: 2 adds per output.
__global__ void pool_line_kernel(const float* __restrict__ in,
                                 float* __restrict__ out,
                                 int stride, int fixmul) {
    int L = blockIdx.x * blockDim.x + threadIdx.x;       // 0 .. 4*160*160-1
    if (L >= DIMB * PLANE) return;
    int w = L % DIML;
    int f = (L / DIML) % DIML;
    int b = L / PLANE;
    int base = b * VOL + f * fixmul + w;
    const float* ip = in + base;
    float* op = out + base;

    float s = 0.0f;
    #pragma unroll
    for (int j = 0; j < KRAD; ++j) s += ip[j * stride];
    for (int h = 0; h < DIML; ++h) {
        if (h + KRAD < DIML) s += ip[(h + KRAD) * stride];
        op[h * stride] = s;
        if (h >= KRAD) s -= ip[(h - KRAD) * stride];
    }
}

// ------------------------------ W-axis pass --------------------------------
// One block = one contiguous 160-float row. Row is staged into LDS with the
// CDNA5 async-to-LDS path (ASYNCcnt), then an inclusive prefix sum in LDS
// turns each 31-tap window into a single subtraction.
__global__ void pool_w_kernel(const float* __restrict__ in,
                              float* __restrict__ out) {
    __shared__ float rowbuf[DIML];
    __shared__ float pre[DIML];
    int tid = threadIdx.x;            // 0..159  (5 waves)
    int base = blockIdx.x * DIML;     // (b,d,h) flat -> contiguous row

    // Async copy: 4 bytes from global address rp + tid*4 into LDS slot rowbuf[tid] (one b32 per lane)
    const float* rp = in + base;                          // uniform -> SGPRs
    unsigned lds_addr = (unsigned)(uintptr_t)(&rowbuf[tid]);
    asm volatile("global_load_async_to_lds_b32 %0, %1, %2"
                 :: "v"(lds_addr), "v"(tid * 4), "s"(rp) : "memory");
    asm volatile("s_wait_asynccnt 0" ::: "memory");
    __syncthreads();

    pre[tid] = rowbuf[tid];
    __syncthreads();
    // Hillis-Steele inclusive prefix over 160 elements (8 steps)
    for (int off = 1; off < DIML; off <<= 1) {
        float add = (tid >= off) ? pre[tid - off] : 0.0f;
        __syncthreads();
        pre[tid] += add;
        __syncthreads();
    }
    int hi = tid + KRAD; if (hi > DIML - 1) hi = DIML - 1;
    float s = pre[hi] - ((tid >= KRAD + 1) ? pre[tid - KRAD - 1] : 0.0f);
    out[base + tid] = s;
}

// ---------------------- fused elementwise + reduction ----------------------
__global__ void fused_reduce_kernel(const float* __restrict__ x,
                                    const float* __restrict__ t,
                                    const float* __restrict__ pool,
                                    float* __restrict__ partials) {
    int blk = blockIdx.x;             // 0 .. 4*512-1
    int b = blk >> 9;
    int c = blk & (CHUNKS_PER_B - 1);
    int b_off = b * VOL;
    int lo = c * CHUNK_ELEMS, hi = lo + CHUNK_ELEMS;

    float sw = 0.f, siw = 0.f, suw = 0.f, si = 0.f, su = 0.f, sb = 0.f;
    for (int j = lo + (int)threadIdx.x; j < hi; j += 256) {
        int idx = b_off + j;
        float xv = x[idx];
        float tv = t[idx];
        float ps = pool[idx] * (1.0f / KVOL);
        float weit = 1.0f + 5.0f * fabsf(ps - tv);
        float bce  = fmaxf(xv, 0.0f) - xv * tv + log1pf(expf(-fabsf(xv)));
        float pr   = 1.0f / (1.0f + expf(-xv));
        float pt = pr * tv, ppt = pr + tv;
        sw  += weit;
        siw += pt * weit;
        suw += ppt * weit;
        si  += pt;
        su  += ppt;
        sb  += bce;
    }

    __shared__ float red[256];
    float vals[NQ] = {sw, siw, suw, si, su, sb};
    for (int q = 0; q < NQ; ++q) {
        red[threadIdx.x] = vals[q];
        __syncthreads();
        for (int offs = 128; offs > 0; offs >>= 1) {
            if ((int)threadIdx.x < offs) red[threadIdx.x] += red[threadIdx.x + offs];
            __syncthreads();
        }
        if (threadIdx.x == 0)
            partials[(b * NQ + q) * CHUNKS_PER_B + c] = red[0];
        __syncthreads();
    }
}

// ------------------------------- finalize ----------------------------------
// One wave (EXEC all ones, required for WMMA). Each of the 24 (b,q) partial
// vectors (512 floats) is summed with 8 chained v_wmma_f32_16x16x4_f32 ops
// against an all-ones B: D = A x 1 + C accumulates 64 values per issue.
// B is all-ones, so lane->(M,K) placement of A is irrelevant to the total.
__global__ void finalize_kernel(const float* __restrict__ partials,
                                float* __restrict__ out) {
    int lane = threadIdx.x;           // exactly 32 threads
    v2f ones; ones.x = 1.0f; ones.y = 1.0f;

    float sums[DIMB * NQ];
    for (int r = 0; r < DIMB * NQ; ++r) {
        const float* p = partials + r * CHUNKS_PER_B;
        v8f acc;
        #pragma unroll
        for (int i = 0; i < 8; ++i) acc[i] = 0.0f;
        #pragma unroll
        for (int ch = 0; ch < 8; ++ch) {
            v2f a;
            a.x = p[ch * 64 + lane];
            a.y = p[ch * 64 + 32 + lane];
            acc = __builtin_amdgcn_wmma_f32_16x16x4_f32(
                false, a, false, ones, (short)0, acc, false, false);
        }
        float s = acc[0] + acc[1] + acc[2] + acc[3]
                + acc[4] + acc[5] + acc[6] + acc[7];
        // D rows M=0..7 live in lanes 0..15, M=8..15 in lanes 16..31.
        sums[r] = __shfl(s, 0, 32) + __shfl(s, 16, 32);
    }

    float bce_total = sums[0 * NQ + 5] + sums[1 * NQ + 5]
                    + sums[2 * NQ + 5] + sums[3 * NQ + 5];
    float wbce_scalar = bce_total / (float)NTOT;

    float acc_loss = 0.f, acc_dice = 0.f, acc_iou = 0.f;
    for (int b = 0; b < DIMB; ++b) {
        float swt = sums[b * NQ + 0];
        float iw  = sums[b * NQ + 1];
        float uw  = sums[b * NQ + 2];
        float i_  = sums[b * NQ + 3] + 0.5e-5f;   // + SMOOTH/2
        float u_  = sums[b * NQ + 4] + 1.0e-5f;   // + SMOOTH
        float wbce = (swt * wbce_scalar) / swt;   // == wbce_scalar (faithful)
        float wiou = 1.0f - (iw + 1.0f) / (uw - iw + 1.0f);
        acc_loss += wbce + wiou;
        acc_dice += 2.0f * i_ / u_;
        acc_iou  += i_ / (u_ - i_);
    }
    if (lane == 0) {
        out[0] = acc_loss * 0.25f;   // total_loss
        out[1] = acc_dice * 0.25f;   // dice.mean(axis=0)  (C==1)
        out[2] = acc_iou  * 0.25f;   // iou.mean(axis=0)
    }
}

// ------------------------------- launcher ----------------------------------
extern "C" void kernel_launch(void* const* d_in, const int* in_sizes, int n_in,
                              void* d_out, int out_size, void* d_ws, size_t ws_size,
                              hipStream_t stream) {
    const float* input  = (const float*)d_in[0];
    const float* target = (const float*)d_in[1];

    float* bufA = (float*)d_ws;             // 64 MB
    float* bufB = bufA + NTOT;              // 64 MB
    float* partials = bufB + NTOT;          // 4*6*512 floats = 48 KB

    int nlines = DIMB * PLANE;              // 102400 lines per strided pass

    // pass 1: pool along H (stride 160, fixed-dim d multiplier 25600)
    pool_line_kernel<<<(nlines + 255) / 256, 256, 0, stream>>>(
        target, bufA, DIML, PLANE);
    // pass 2: pool along D (stride 25600, fixed-dim h multiplier 160)
    pool_line_kernel<<<(nlines + 255) / 256, 256, 0, stream>>>(
        bufA, bufB, PLANE, DIML);
    // pass 3: pool along W (contiguous rows, async->LDS + prefix sums)
    pool_w_kernel<<<DIMB * PLANE, DIML, 0, stream>>>(bufB, bufA);
    // pass 4: fused elementwise + partial reductions
    fused_reduce_kernel<<<DIMB * CHUNKS_PER_B, 256, 0, stream>>>(
        input, target, bufA, partials);
    // pass 5: WMMA tree-sum of partials + scalar outputs
    finalize_kernel<<<1, 32, 0, stream>>>(partials, (float*)d_out);
}